// GCN_67774583931478
// MI455X (gfx1250) — compile-verified
//
#include <hip/hip_runtime.h>
#include <hip/hip_bf16.h>

// GraphSAGE 2-layer forward for MI455X (gfx1250, wave32).
//   h   = (S(x)/max(cnt,1)) @ Wl1^T + bl1 + x @ Wr1^T
//   out = (S(h)/max(cnt,1)) @ Wl2^T + bl2 + h @ Wr2^T
// where S(v)[n] = sum over edges e with dst[e]==n of v[src[e]].
//
// Working set (~77 MB of node matrices) fits in the 192 MB L2, so the
// edge scatter (the dominant cost) runs out of L2 with non-returning
// global_atomic_add_f32. GEMMs use V_WMMA_F32_16X16X4_F32 (full fp32).

#define SAGE_N 50000
#define SAGE_D 128

typedef __attribute__((ext_vector_type(2))) float v2f;
typedef __attribute__((ext_vector_type(8))) float v8f;

__global__ void sage_zero_kernel(float* __restrict__ p, size_t n) {
    size_t i = (size_t)blockIdx.x * blockDim.x + threadIdx.x;
    size_t stride = (size_t)gridDim.x * blockDim.x;
    for (; i < n; i += stride) p[i] = 0.0f;
}

__global__ void sage_count_kernel(const int* __restrict__ dst,
                                  float* __restrict__ cnt, int E) {
    int e = blockIdx.x * blockDim.x + threadIdx.x;
    if (e < E) unsafeAtomicAdd(&cnt[dst[e]], 1.0f);
}

// One wave (32 lanes) per edge; each lane moves 4 contiguous floats.
__global__ __launch_bounds__(256) void sage_scatter_kernel(
    const float* __restrict__ xin, const int* __restrict__ src,
    const int* __restrict__ dst, float* __restrict__ s, int E) {
    int e = blockIdx.x * 8 + (threadIdx.x >> 5);
    if (e >= E) return;
    int lane = threadIdx.x & 31;
    int sn = src[e];
    int dn = dst[e];
    const float4 v = *(const float4*)(xin + (size_t)sn * SAGE_D + lane * 4);
    float* sd = s + (size_t)dn * SAGE_D + lane * 4;
    unsafeAtomicAdd(sd + 0, v.x);
    unsafeAtomicAdd(sd + 1, v.y);
    unsafeAtomicAdd(sd + 2, v.z);
    unsafeAtomicAdd(sd + 3, v.w);
}

// Fused: agg = s/max(cnt,1);  out = agg @ Wl^T + bl + x @ Wr^T
// One wave computes a 16-row x 128-col output tile (8 col-tiles of 16x16),
// so each node row of s and x is read exactly once.
// V_WMMA_F32_16X16X4_F32 fragment layouts (ISA 7.12.2), lane = 16*half + lm:
//   A[16x4]:  a[i] = A[lm][2*half+i]           (contiguous float2 of a row)
//   B[4x16]:  b[i] = B[2*half+i][lm] = W[lm][k] (B = W^T -> row of W)
//   C[16x16]: c[v] = C[v+8*half][lm]
__global__ __launch_bounds__(256) void sage_gemm_kernel(
    const float* __restrict__ s,    // [N,128] neighbor sums
    const float* __restrict__ cnt,  // [N]
    const float* __restrict__ xin,  // [N,128] self features
    const float* __restrict__ Wl,   // [128,128]
    const float* __restrict__ bl,   // [128]
    const float* __restrict__ Wr,   // [128,128]
    float* __restrict__ out,        // [N,128]
    int nRowTiles) {
    const int wave = threadIdx.x >> 5;
    const int tile = blockIdx.x * 8 + wave;
    if (tile >= nRowTiles) return;

    const int lane = threadIdx.x & 31;
    const int half = lane >> 4;
    const int lm   = lane & 15;
    const int m0   = tile * 16;

    const float inv = 1.0f / fmaxf(cnt[m0 + lm], 1.0f);

    const float* arow_s = s   + (size_t)(m0 + lm) * SAGE_D + 2 * half;
    const float* arow_x = xin + (size_t)(m0 + lm) * SAGE_D + 2 * half;
    const float* wl_row = Wl  + (size_t)lm * SAGE_D + 2 * half;
    const float* wr_row = Wr  + (size_t)lm * SAGE_D + 2 * half;

    v8f acc[8];
#pragma unroll
    for (int ct = 0; ct < 8; ++ct) acc[ct] = (v8f){};

    for (int k0 = 0; k0 < SAGE_D; k0 += 4) {
        v2f a_agg = *(const v2f*)(arow_s + k0);
        a_agg = a_agg * inv;
        v2f a_x = *(const v2f*)(arow_x + k0);
#pragma unroll
        for (int ct = 0; ct < 8; ++ct) {
            v2f b_l = *(const v2f*)(wl_row + (size_t)ct * 16 * SAGE_D + k0);
            v2f b_r = *(const v2f*)(wr_row + (size_t)ct * 16 * SAGE_D + k0);
            acc[ct] = __builtin_amdgcn_wmma_f32_16x16x4_f32(
                false, a_agg, false, b_l, (short)0, acc[ct], false, false);
            acc[ct] = __builtin_amdgcn_wmma_f32_16x16x4_f32(
                false, a_x, false, b_r, (short)0, acc[ct], false, false);
        }
    }

#pragma unroll
    for (int ct = 0; ct < 8; ++ct) {
        const float bias = bl[ct * 16 + lm];
#pragma unroll
        for (int v = 0; v < 8; ++v) {
            out[(size_t)(m0 + v + 8 * half) * SAGE_D + ct * 16 + lm] =
                acc[ct][v] + bias;
        }
    }
}

extern "C" void kernel_launch(void* const* d_in, const int* in_sizes, int n_in,
                              void* d_out, int out_size, void* d_ws, size_t ws_size,
                              hipStream_t stream) {
    const float* x   = (const float*)d_in[0];
    const int*   ei  = (const int*)d_in[1];
    const float* Wl1 = (const float*)d_in[2];
    const float* bl1 = (const float*)d_in[3];
    const float* Wr1 = (const float*)d_in[4];
    const float* Wl2 = (const float*)d_in[5];
    const float* bl2 = (const float*)d_in[6];
    const float* Wr2 = (const float*)d_in[7];
    float* out = (float*)d_out;

    const int N = SAGE_N;
    const int E = in_sizes[1] / 2;      // edge_index is [2, E]
    const int* src = ei;
    const int* dst = ei + E;

    // Workspace: s [N*D] floats, then cnt [N] floats (~25.8 MB).
    float* s   = (float*)d_ws;
    float* cnt = s + (size_t)N * SAGE_D;

    const int nRowTiles  = N / 16;                  // 3125, exact
    const int gemmBlocks = (nRowTiles + 7) / 8;     // 8 waves per block
    const int scatBlocks = (E + 7) / 8;             // 8 edges per block

    // Degree counts (reused by both layers) + zero s.
    sage_zero_kernel<<<2048, 256, 0, stream>>>(s, (size_t)N * SAGE_D + N);
    sage_count_kernel<<<(E + 255) / 256, 256, 0, stream>>>(dst, cnt, E);

    // Layer 1: h -> d_out
    sage_scatter_kernel<<<scatBlocks, 256, 0, stream>>>(x, src, dst, s, E);
    sage_gemm_kernel<<<gemmBlocks, 256, 0, stream>>>(s, cnt, x, Wl1, bl1, Wr1,
                                                     out, nRowTiles);

    // Layer 2: reads h from d_out, writes final result in place (each wave
    // only reads the 16 rows it itself overwrites, loads precede stores).
    sage_zero_kernel<<<2048, 256, 0, stream>>>(s, (size_t)N * SAGE_D);
    sage_scatter_kernel<<<scatBlocks, 256, 0, stream>>>(out, src, dst, s, E);
    sage_gemm_kernel<<<gemmBlocks, 256, 0, stream>>>(s, cnt, out, Wl2, bl2, Wr2,
                                                     out, nRowTiles);
}